// GINPeptideStruct_20461224198769
// MI455X (gfx1250) — compile-verified
//
#include <hip/hip_runtime.h>
#include <hip/hip_bf16.h>

typedef __attribute__((ext_vector_type(2))) float v2f;
typedef __attribute__((ext_vector_type(8))) float v8f;

// ---------------------------------------------------------------------------
// Layer 1: agg (N x 9, already = x + sum_j x_j) @ W1^T + b1, relu -> h1 (N x 64)
// K=9 is too skinny for WMMA; plain VALU, one thread per (node, out).
// ---------------------------------------------------------------------------
__global__ __launch_bounds__(256)
void gin_lin9_relu(const float* __restrict__ agg, const float* __restrict__ W,
                   const float* __restrict__ bias, float* __restrict__ out, int N) {
    long long idx = (long long)blockIdx.x * blockDim.x + threadIdx.x;
    if (idx >= (long long)N * 64) return;
    int n = (int)(idx >> 6);
    int o = (int)(idx & 63);
    const float* a = agg + (size_t)n * 9;
    const float* w = W + (size_t)o * 9;
    float s = bias[o];
#pragma unroll
    for (int k = 0; k < 9; ++k) s = fmaf(a[k], w[k], s);
    out[idx] = fmaxf(s, 0.0f);
}

// ---------------------------------------------------------------------------
// WMMA GEMM: out[N,C] = relu(A[N,K] @ W[C,K]^T + bias), fp32 via
// V_WMMA_F32_16X16X4_F32. One block = 16 rows; one wave per 16-col tile.
// A tile staged in LDS with padded stride (K+4) to avoid bank conflicts.
// ---------------------------------------------------------------------------
template <int K, int C>
__global__ __launch_bounds__((C / 16) * 32)
void gin_gemm_relu(const float* __restrict__ A, const float* __restrict__ W,
                   const float* __restrict__ bias, float* __restrict__ out, int N) {
    constexpr int WAVES = C / 16;
    constexpr int LDA = K + 4;
    __shared__ float As[16 * LDA];

    const int rowBase = blockIdx.x * 16;
    const int tid = threadIdx.x;

    // Cooperative, coalesced load of the 16 x K A tile into LDS.
    for (int i = tid; i < 16 * K; i += WAVES * 32) {
        int r = i / K, c = i % K;
        int row = rowBase + r;
        As[r * LDA + c] = (row < N) ? A[(size_t)row * K + c] : 0.0f;
    }
    __syncthreads();

    const int wave = tid >> 5;
    const int lane = tid & 31;
    const int colBase = wave * 16;
    const int lcol = lane & 15;           // M for A-frag, N (output col) for B/C/D
    const int khalf = (lane >> 4) << 1;   // lanes 0-15 -> K+{0,1}; 16-31 -> K+{2,3}

    v8f acc = {0.f, 0.f, 0.f, 0.f, 0.f, 0.f, 0.f, 0.f};
    const float* wrow = W + (size_t)(colBase + lcol) * K;   // B[k][col] = W[col][k]
    const float* arow = As + lcol * LDA;

#pragma unroll
    for (int k = 0; k < K; k += 4) {
        v2f a, b;
        a.x = arow[k + khalf];
        a.y = arow[k + khalf + 1];
        b.x = wrow[k + khalf];
        b.y = wrow[k + khalf + 1];
        // 8 args: (neg_a, A, neg_b, B, c_mod, C, reuse_a, reuse_b)
        acc = __builtin_amdgcn_wmma_f32_16x16x4_f32(false, a, false, b,
                                                    (short)0, acc, false, false);
    }

    const float bv = bias[colBase + lcol];
    const int mBase = (lane >> 4) << 3;   // upper half-wave holds rows 8..15
#pragma unroll
    for (int v = 0; v < 8; ++v) {
        int row = rowBase + mBase + v;
        if (row < N) {
            float val = acc[v] + bv;
            out[(size_t)row * C + colBase + lcol] = fmaxf(val, 0.0f);
        }
    }
}

// ---------------------------------------------------------------------------
// Edge scatter-add: agg[dst] += h[src]. agg pre-initialized with h (self term).
// ---------------------------------------------------------------------------
__global__ __launch_bounds__(256)
void scatter_edges9(const float* __restrict__ h, const int* __restrict__ src,
                    const int* __restrict__ dst, float* __restrict__ agg, int E) {
    int e = blockIdx.x * blockDim.x + threadIdx.x;
    if (e >= E) return;
    int s = src[e], d = dst[e];
    const float* hp = h + (size_t)s * 9;
    float* ap = agg + (size_t)d * 9;
#pragma unroll
    for (int k = 0; k < 9; ++k) atomicAdd(ap + k, hp[k]);
}

template <int CD4>  // C/4 vec4 chunks per edge
__global__ __launch_bounds__(256)
void scatter_edges_vec4(const float* __restrict__ h, const int* __restrict__ src,
                        const int* __restrict__ dst, float* __restrict__ agg, int E) {
    long long idx = (long long)blockIdx.x * blockDim.x + threadIdx.x;
    if (idx >= (long long)E * CD4) return;
    int e = (int)(idx / CD4);
    int c4 = (int)(idx % CD4) * 4;
    constexpr int C = CD4 * 4;
    int s = src[e], d = dst[e];
    float4 v = *(const float4*)(h + (size_t)s * C + c4);
    float* ap = agg + (size_t)d * C + c4;
    atomicAdd(ap + 0, v.x);
    atomicAdd(ap + 1, v.y);
    atomicAdd(ap + 2, v.z);
    atomicAdd(ap + 3, v.w);
}

// ---------------------------------------------------------------------------
// Pooling: per-graph feature sums and node counts (batch ids sorted, 512 graphs)
// ---------------------------------------------------------------------------
__global__ __launch_bounds__(256)
void pool_sum(const float* __restrict__ h, const int* __restrict__ batch,
              float* __restrict__ sums, int N) {
    long long idx = (long long)blockIdx.x * blockDim.x + threadIdx.x;
    if (idx >= (long long)N * 64) return;
    int n = (int)(idx >> 6);
    int f = (int)(idx & 63);
    atomicAdd(&sums[(size_t)batch[n] * 64 + f], h[idx]);
}

__global__ __launch_bounds__(256)
void pool_count(const int* __restrict__ batch, float* __restrict__ counts, int N) {
    int n = blockIdx.x * blockDim.x + threadIdx.x;
    if (n >= N) return;
    atomicAdd(&counts[batch[n]], 1.0f);
}

// ---------------------------------------------------------------------------
// Final FC: out[512,11] = (sums/max(counts,1)) @ Wfc^T + bfc
// ---------------------------------------------------------------------------
__global__ __launch_bounds__(256)
void fc_out_kernel(const float* __restrict__ sums, const float* __restrict__ counts,
                   const float* __restrict__ Wfc, const float* __restrict__ bfc,
                   float* __restrict__ out) {
    int idx = blockIdx.x * blockDim.x + threadIdx.x;
    if (idx >= 512 * 11) return;
    int g = idx / 11;
    int c = idx % 11;
    float inv = 1.0f / fmaxf(counts[g], 1.0f);
    const float* s = sums + (size_t)g * 64;
    const float* w = Wfc + (size_t)c * 64;
    float acc = bfc[c];
#pragma unroll
    for (int k = 0; k < 64; ++k) acc = fmaf(s[k] * inv, w[k], acc);
    out[idx] = acc;
}

// ---------------------------------------------------------------------------
// Host-side orchestration
// ---------------------------------------------------------------------------
extern "C" void kernel_launch(void* const* d_in, const int* in_sizes, int n_in,
                              void* d_out, int out_size, void* d_ws, size_t ws_size,
                              hipStream_t stream) {
    const float* x   = (const float*)d_in[0];   // [N,9]
    const int*   ei  = (const int*)d_in[1];     // [2,E]
    const int*   bat = (const int*)d_in[2];     // [N]
    const float* W1  = (const float*)d_in[3];   // [64,9]
    const float* b1  = (const float*)d_in[4];
    const float* W2  = (const float*)d_in[5];   // [128,64]
    const float* b2  = (const float*)d_in[6];
    const float* W3  = (const float*)d_in[7];   // [64,128]
    const float* b3  = (const float*)d_in[8];
    const float* Wfc = (const float*)d_in[9];   // [11,64]
    const float* bfc = (const float*)d_in[10];
    float* out = (float*)d_out;                 // [512,11]

    const int N = in_sizes[0] / 9;
    const int E = in_sizes[1] / 2;
    const int* src = ei;
    const int* dst = ei + E;

    // Workspace layout (floats)
    float* ws = (float*)d_ws;
    float* agg    = ws;                                   // N*128 (reused per layer)
    float* h2     = agg + (size_t)N * 128;                // N*128
    float* hA     = h2 + (size_t)N * 128;                 // N*64 (h1, then h3)
    float* sums   = hA + (size_t)N * 64;                  // 512*64
    float* counts = sums + 512 * 64;                      // 512

    const int T = 256;
    auto nb = [](long long work, int t) { return (unsigned)((work + t - 1) / t); };

    // ---- Layer 1: agg = x + scatter(x); h1 = relu(agg @ W1^T + b1) ----
    hipMemcpyAsync(agg, x, (size_t)N * 9 * sizeof(float), hipMemcpyDeviceToDevice, stream);
    scatter_edges9<<<nb(E, T), T, 0, stream>>>(x, src, dst, agg, E);
    gin_lin9_relu<<<nb((long long)N * 64, T), T, 0, stream>>>(agg, W1, b1, hA, N);

    // ---- Layer 2: agg = h1 + scatter(h1); h2 = relu(agg @ W2^T + b2) ----
    hipMemcpyAsync(agg, hA, (size_t)N * 64 * sizeof(float), hipMemcpyDeviceToDevice, stream);
    scatter_edges_vec4<16><<<nb((long long)E * 16, T), T, 0, stream>>>(hA, src, dst, agg, E);
    gin_gemm_relu<64, 128><<<nb(N, 16), 256, 0, stream>>>(agg, W2, b2, h2, N);

    // ---- Layer 3: agg = h2 + scatter(h2); h3 = relu(agg @ W3^T + b3) ----
    hipMemcpyAsync(agg, h2, (size_t)N * 128 * sizeof(float), hipMemcpyDeviceToDevice, stream);
    scatter_edges_vec4<32><<<nb((long long)E * 32, T), T, 0, stream>>>(h2, src, dst, agg, E);
    gin_gemm_relu<128, 64><<<nb(N, 16), 128, 0, stream>>>(agg, W3, b3, hA, N);  // h3 -> hA

    // ---- Mean pool + FC ----
    hipMemsetAsync(sums, 0, (512 * 64 + 512) * sizeof(float), stream);
    pool_sum<<<nb((long long)N * 64, T), T, 0, stream>>>(hA, bat, sums, N);
    pool_count<<<nb(N, T), T, 0, stream>>>(bat, counts, N);
    fc_out_kernel<<<nb(512 * 11, T), T, 0, stream>>>(sums, counts, Wfc, bfc, out);
}